// Decoder_50440095924346
// MI455X (gfx1250) — compile-verified
//
#include <hip/hip_runtime.h>
#include <hip/hip_bf16.h>

// ---------------- problem constants ----------------
#define HID   64
#define LAT   128
#define OUTC  3
#define N0    6250
#define N1    25000
#define N2    100000
#define E0    100000
#define E1    400000
#define E2    1600000
#define KNN   16

typedef __attribute__((ext_vector_type(2))) float v2f;
typedef __attribute__((ext_vector_type(8))) float v8f;

#define TSTRIDE 68   // padded LDS tile row stride (floats); 68*4B = 272B ≡ 0 mod 16

// ---------------- WMMA helper (fp32 16x16x4) ----------------
__device__ inline v8f wmma4(v2f a, v2f b, v8f c) {
    // D = A(16x4 f32) x B(4x16 f32) + C(16x16 f32)
    return __builtin_amdgcn_wmma_f32_16x16x4_f32(
        /*neg_a=*/false, a, /*neg_b=*/false, b,
        /*c_mod=*/(short)0, c, /*reuse_a=*/false, /*reuse_b=*/false);
}

// acc[jt] += bias broadcast along columns (C/D layout: col = 16*jt + (lane&15))
__device__ inline void apply_bias(v8f acc[4], const float* __restrict__ b, int lane) {
    int m = lane & 15;
#pragma unroll
    for (int jt = 0; jt < 4; ++jt) {
        float bj = b[jt * 16 + m];
#pragma unroll
        for (int v = 0; v < 8; ++v) acc[jt][v] += bj;
    }
}

// store C/D-layout accumulators into the padded 16xTSTRIDE LDS tile
__device__ inline void store_tile(float* __restrict__ tile, int lane,
                                  const v8f acc[4], bool relu) {
    int m  = lane & 15;
    int rb = (lane >> 4) * 8;       // M = v + rb
#pragma unroll
    for (int jt = 0; jt < 4; ++jt) {
#pragma unroll
        for (int v = 0; v < 8; ++v) {
            float val = acc[jt][v];
            if (relu) val = fmaxf(val, 0.0f);
            tile[(rb + v) * TSTRIDE + jt * 16 + m] = val;
        }
    }
}

// GEMM: acc[jt] += Xtile(16x64, LDS) * W^T ; W is [64][64] row-major (out,in)
// Branch-free: A fragments from LDS (ds_load_b64), B fragments from global (b64).
__device__ inline void gemm_lds(const float* __restrict__ tile, int lane,
                                const float* __restrict__ W, v8f acc[4]) {
    int m  = lane & 15;             // A: row M ; B: col N (same lane math)
    int kh = (lane >> 4) * 2;       // K sub-pair selected by lane half
#pragma unroll
    for (int k0 = 0; k0 < 64; k0 += 4) {
        v2f a = *(const v2f*)(tile + m * TSTRIDE + k0 + kh);   // 8B aligned
#pragma unroll
        for (int jt = 0; jt < 4; ++jt) {
            // B[k][n] = W[n][k]
            v2f b = *(const v2f*)(W + (jt * 16 + m) * 64 + k0 + kh);
            acc[jt] = wmma4(a, b, acc[jt]);
        }
    }
}

// cooperative, coalesced load of a 16x64 node tile into LDS (zero-fill N-tail)
__device__ inline void load_tile_global(float* __restrict__ tile,
                                        const float* __restrict__ x,
                                        long nodebase, int N, int lane) {
    int row  = lane >> 1;            // 2 lanes per row
    int half = (lane & 1) * 32;      // 32 floats each
    long r = nodebase + row;
    float4* dst = (float4*)(tile + row * TSTRIDE + half);
    if (r < (long)N) {
        const float4* src = (const float4*)(x + r * 64 + half);
#pragma unroll
        for (int q = 0; q < 8; ++q) dst[q] = src[q];
    } else {
        float4 z4 = {0.f, 0.f, 0.f, 0.f};
#pragma unroll
        for (int q = 0; q < 8; ++q) dst[q] = z4;
    }
}

// cooperative, coalesced write-back of the tile to global (guarded rows)
__device__ inline void store_tile_out(const float* __restrict__ tile,
                                      float* __restrict__ out,
                                      long nodebase, int N, int lane) {
    int row  = lane >> 1;
    int half = (lane & 1) * 32;
    long r = nodebase + row;
    if (r < (long)N) {
        const float4* src = (const float4*)(tile + row * TSTRIDE + half);
        float4* dst = (float4*)(out + r * 64 + half);
#pragma unroll
        for (int q = 0; q < 8; ++q) dst[q] = src[q];
    }
}

// ---------------- GIN MLP, 64->64->64->64->64 with skip (convs 0..3) ----------------
// 256 threads = 8 waves; each wave owns a 16-node tile and a private LDS staging tile.
__global__ __launch_bounds__(256) void gin_mlp64_kernel(
    const float* __restrict__ yin,  // [N,64] = x + agg
    float* __restrict__ out,        // [N,64]
    const float* __restrict__ W0, const float* __restrict__ b0,
    const float* __restrict__ W1, const float* __restrict__ b1,
    const float* __restrict__ W2, const float* __restrict__ b2,
    const float* __restrict__ W3, const float* __restrict__ b3,
    int N, int relu_out)
{
    __shared__ float sh[8 * 16 * TSTRIDE];
    int lane = threadIdx.x & 31;
    int wave = threadIdx.x >> 5;
    float* tile = sh + wave * (16 * TSTRIDE);
    long nodebase = ((long)blockIdx.x * 8 + wave) * 16;
    if (nodebase >= (long)N) return;   // wave-uniform exit; no block-level syncs used

    v8f z = {0.f,0.f,0.f,0.f,0.f,0.f,0.f,0.f};

    // stage input tile (coalesced b128), then run all layers branch-free
    load_tile_global(tile, yin, nodebase, N, lane);

    // layer 0: t0 = relu(X @ W0^T + b0)
    v8f acc[4] = {z, z, z, z};
    gemm_lds(tile, lane, W0, acc);
    apply_bias(acc, b0, lane);
    store_tile(tile, lane, acc, /*relu=*/true);

    // layer 1: o1 = t0 @ W1^T + b1  (kept in registers for the skip)
    v8f o1[4] = {z, z, z, z};
    gemm_lds(tile, lane, W1, o1);
    apply_bias(o1, b1, lane);
    store_tile(tile, lane, o1, /*relu=*/false);   // tile <- o1

    // layer 2: t2 = relu(o1 @ W2^T + b2)
    v8f acc2[4] = {z, z, z, z};
    gemm_lds(tile, lane, W2, acc2);
    apply_bias(acc2, b2, lane);
    store_tile(tile, lane, acc2, /*relu=*/true);  // tile <- t2

    // layer 3: out = t2 @ W3^T + b3 + o1   (o1 rides in as the C operand)
    v8f acc3[4] = {o1[0], o1[1], o1[2], o1[3]};
    gemm_lds(tile, lane, W3, acc3);
    apply_bias(acc3, b3, lane);
    store_tile(tile, lane, acc3, relu_out != 0);  // tile <- result
    store_tile_out(tile, out, nodebase, N, lane); // coalesced b128 write-back
}

// ---------------- final GIN MLP, 64->64->3->64->3 with skip (conv 4) ----------------
__global__ __launch_bounds__(256) void gin_mlp_final_kernel(
    const float* __restrict__ y, float* __restrict__ out,
    const float* __restrict__ W0, const float* __restrict__ b0,
    const float* __restrict__ W1, const float* __restrict__ b1,
    const float* __restrict__ W2, const float* __restrict__ b2,
    const float* __restrict__ W3, const float* __restrict__ b3, int N)
{
    __shared__ float sW0[64 * 64], sb0[64];
    __shared__ float sW1[3 * 64],  sb1[3];
    __shared__ float sW2[64 * 3],  sb2[64];
    __shared__ float sW3[3 * 64],  sb3[3];
    int t = threadIdx.x;
    for (int i = t; i < 64 * 64; i += 256) sW0[i] = W0[i];
    for (int i = t; i < 192; i += 256) { sW1[i] = W1[i]; sW2[i] = W2[i]; sW3[i] = W3[i]; }
    if (t < 64) { sb0[t] = b0[t]; sb2[t] = b2[t]; }
    if (t < 3)  { sb1[t] = b1[t]; sb3[t] = b3[t]; }
    __syncthreads();

    long i = (long)blockIdx.x * blockDim.x + t;
    if (i >= (long)N) return;

    float xr[64];
#pragma unroll
    for (int c = 0; c < 64; ++c) xr[c] = y[i * 64 + c];

    float o1[3] = {sb1[0], sb1[1], sb1[2]};
    for (int j = 0; j < 64; ++j) {
        float acc = sb0[j];
#pragma unroll
        for (int c = 0; c < 64; ++c) acc += sW0[j * 64 + c] * xr[c];
        float tv = fmaxf(acc, 0.0f);
        o1[0] += sW1[0 * 64 + j] * tv;
        o1[1] += sW1[1 * 64 + j] * tv;
        o1[2] += sW1[2 * 64 + j] * tv;
    }
    float o3[3] = {sb3[0] + o1[0], sb3[1] + o1[1], sb3[2] + o1[2]};
    for (int j = 0; j < 64; ++j) {
        float tv = fmaxf(sW2[j * 3 + 0] * o1[0] + sW2[j * 3 + 1] * o1[1]
                       + sW2[j * 3 + 2] * o1[2] + sb2[j], 0.0f);
        o3[0] += sW3[0 * 64 + j] * tv;
        o3[1] += sW3[1 * 64 + j] * tv;
        o3[2] += sW3[2 * 64 + j] * tv;
    }
    out[i * 3 + 0] = o3[0];
    out[i * 3 + 1] = o3[1];
    out[i * 3 + 2] = o3[2];
}

// ---------------- head: h = relu(W0 @ latent + b0)  (128x128) ----------------
__global__ void lin0_kernel(const float* __restrict__ latent,
                            const float* __restrict__ w, const float* __restrict__ b,
                            float* __restrict__ h)
{
    __shared__ float sl[LAT];
    int t = threadIdx.x;
    sl[t] = latent[t];
    __syncthreads();
    float acc = b[t];
#pragma unroll 8
    for (int k = 0; k < LAT; ++k) acc += w[t * LAT + k] * sl[k];
    h[t] = fmaxf(acc, 0.0f);
}

// ---------------- head: x = relu(W1 @ h + b1)  (400000x128, streaming) --------
__global__ void lin1_kernel(const float* __restrict__ w, const float* __restrict__ b,
                            const float* __restrict__ h, float* __restrict__ x, int M)
{
    __shared__ float sh[LAT];
    if (threadIdx.x < LAT) sh[threadIdx.x] = h[threadIdx.x];
    __syncthreads();
    long i = (long)blockIdx.x * blockDim.x + threadIdx.x;
    if (i >= (long)M) return;
    const float* wr = w + i * LAT;
    float acc = b[i];
#pragma unroll 8
    for (int k = 0; k < LAT; ++k) acc += wr[k] * sh[k];
    x[i] = fmaxf(acc, 0.0f);
}

// ---------------- utility: copy ----------------
__global__ void copy_kernel(const float* __restrict__ src, float* __restrict__ dst, long n)
{
    long i = (long)blockIdx.x * blockDim.x + threadIdx.x;
    if (i < n) dst[i] = src[i];
}

// ---------------- GIN aggregation: y[dst] += x[src] over edges ----------------
__global__ void edge_scatter_kernel(const float* __restrict__ x,
                                    const int* __restrict__ ei,
                                    float* __restrict__ y, int E)
{
    long t = (long)blockIdx.x * blockDim.x + threadIdx.x;
    if (t >= (long)E * 16) return;
    int e  = (int)(t >> 4);
    int c4 = (int)(t & 15) << 2;
    int src = ei[e];        // row 0: gather sources
    int dst = ei[E + e];    // row 1: segment ids
    const float4 v = *(const float4*)(x + (long)src * 64 + c4);
    atomicAdd(&y[(long)dst * 64 + c4 + 0], v.x);
    atomicAdd(&y[(long)dst * 64 + c4 + 1], v.y);
    atomicAdd(&y[(long)dst * 64 + c4 + 2], v.z);
    atomicAdd(&y[(long)dst * 64 + c4 + 3], v.w);
}

// ---------------- brute-force kNN (k=16), sorted register list ----------------
__global__ void knn_topk_kernel(const float* __restrict__ pos_y, int Ny,
                                const float* __restrict__ pos_x, int Nx,
                                int* __restrict__ idx_out, float* __restrict__ w_out)
{
    long i = (long)blockIdx.x * blockDim.x + threadIdx.x;
    if (i >= (long)Ny) return;
    float yx = pos_y[i * 3 + 0], yy = pos_y[i * 3 + 1], yz = pos_y[i * 3 + 2];

    float bd[KNN];
    int   bi[KNN];
#pragma unroll
    for (int s = 0; s < KNN; ++s) { bd[s] = 3.4e38f; bi[s] = 0; }

    for (int j = 0; j < Nx; ++j) {
        float dx = pos_x[j * 3 + 0] - yx;
        float dy = pos_x[j * 3 + 1] - yy;
        float dz = pos_x[j * 3 + 2] - yz;
        float d2 = dx * dx + dy * dy + dz * dz;
        if (d2 < bd[KNN - 1]) {
            bd[KNN - 1] = d2; bi[KNN - 1] = j;
#pragma unroll
            for (int s = KNN - 1; s > 0; --s) {
                if (bd[s] < bd[s - 1]) {
                    float td = bd[s]; bd[s] = bd[s - 1]; bd[s - 1] = td;
                    int   ti = bi[s]; bi[s] = bi[s - 1]; bi[s - 1] = ti;
                }
            }
        }
    }
    float w[KNN], wsum = 0.0f;
#pragma unroll
    for (int s = 0; s < KNN; ++s) {
        float d = fmaxf(bd[s], 0.0f);
        d = fmaxf(d, 1e-16f);
        w[s] = 1.0f / d;
        wsum += w[s];
    }
    float inv = 1.0f / wsum;
#pragma unroll
    for (int s = 0; s < KNN; ++s) {
        idx_out[i * KNN + s] = bi[s];
        w_out[i * KNN + s]   = w[s] * inv;   // pre-normalized
    }
}

// ---------------- kNN weighted gather: out[y,c] = sum_s w*x[idx,c] ----------------
__global__ void interp_gather_kernel(const float* __restrict__ xsrc,
                                     const int* __restrict__ idx,
                                     const float* __restrict__ w,
                                     float* __restrict__ out, int Ny)
{
    long t = (long)blockIdx.x * blockDim.x + threadIdx.x;
    if (t >= (long)Ny * 64) return;
    int i = (int)(t >> 6);
    int c = (int)(t & 63);
    float acc = 0.0f;
#pragma unroll
    for (int s = 0; s < KNN; ++s)
        acc += w[i * KNN + s] * xsrc[(long)idx[i * KNN + s] * 64 + c];
    out[(long)i * 64 + c] = acc;
}

// ---------------- host orchestration ----------------
static inline int cdivl(long a, long b) { return (int)((a + b - 1) / b); }

extern "C" void kernel_launch(void* const* d_in, const int* in_sizes, int n_in,
                              void* d_out, int out_size, void* d_ws, size_t ws_size,
                              hipStream_t stream)
{
    (void)n_in; (void)out_size; (void)ws_size;

    const float* latent = (const float*)d_in[0];
    const float* pos0   = (const float*)d_in[1];
    const float* pos1   = (const float*)d_in[2];
    const float* pos2   = (const float*)d_in[3];

    // params pytree order: detect whether 'lin0' leaves or 'convs' leaves come first
    const float* lin0w; const float* lin0b; const float* lin1w; const float* lin1b;
    const float* conv[5][8];
    int edgeBase = 48;   // 4 non-param inputs + 44 param leaves, either order
    if (in_sizes[4] == LAT * LAT) {
        // insertion order: lin0(w,b), lin1(w,b), convs[5]{l0..l3 (w,b)}
        lin0w = (const float*)d_in[4]; lin0b = (const float*)d_in[5];
        lin1w = (const float*)d_in[6]; lin1b = (const float*)d_in[7];
        for (int i = 0; i < 5; ++i)
            for (int j = 0; j < 8; ++j)
                conv[i][j] = (const float*)d_in[8 + i * 8 + j];
    } else {
        // sorted-key pytree order: convs first, then lin0, lin1
        for (int i = 0; i < 5; ++i)
            for (int j = 0; j < 8; ++j)
                conv[i][j] = (const float*)d_in[4 + i * 8 + j];
        lin0w = (const float*)d_in[44]; lin0b = (const float*)d_in[45];
        lin1w = (const float*)d_in[46]; lin1b = (const float*)d_in[47];
    }
    const int* e0 = (const int*)d_in[edgeBase + 0];
    const int* e1 = (const int*)d_in[edgeBase + 1];
    const int* e2 = (const int*)d_in[edgeBase + 2];

    // workspace layout
    float* X  = (float*)d_ws;                   // [N2*64] current features
    float* T  = X + (size_t)N2 * 64;            // [N2*64] scratch (agg / interp)
    int*   KI = (int*)(T + (size_t)N2 * 64);    // [N2*16] knn indices
    float* KW = (float*)(KI + (size_t)N2 * 16); // [N2*16] normalized knn weights
    float* H  = KW + (size_t)N2 * 16;           // [128]

    const int TPB = 256;

    // head
    lin0_kernel<<<1, LAT, 0, stream>>>(latent, lin0w, lin0b, H);
    lin1_kernel<<<cdivl((long)N0 * HID, TPB), TPB, 0, stream>>>(lin1w, lin1b, H, X, N0 * HID);

    // conv64: X <- relu(GIN(X)) using buffers X (in/out) and T (x + agg)
    auto conv64 = [&](const float* const* cp, const int* ei, int E, int N, int relu) {
        long n = (long)N * 64;
        copy_kernel<<<cdivl(n, TPB), TPB, 0, stream>>>(X, T, n);
        edge_scatter_kernel<<<cdivl((long)E * 16, TPB), TPB, 0, stream>>>(X, ei, T, E);
        gin_mlp64_kernel<<<cdivl(N, 128), 256, 0, stream>>>(
            T, X, cp[0], cp[1], cp[2], cp[3], cp[4], cp[5], cp[6], cp[7], N, relu);
    };

    conv64(conv[0], e0, E0, N0, 1);
    conv64(conv[1], e0, E0, N0, 1);

    // pool 0: interpolate N0 -> N1, then conv[2]
    knn_topk_kernel<<<cdivl(N1, TPB), TPB, 0, stream>>>(pos1, N1, pos0, N0, KI, KW);
    interp_gather_kernel<<<cdivl((long)N1 * 64, TPB), TPB, 0, stream>>>(X, KI, KW, T, N1);
    copy_kernel<<<cdivl((long)N1 * 64, TPB), TPB, 0, stream>>>(T, X, (long)N1 * 64);
    conv64(conv[2], e1, E1, N1, 1);

    // pool 1: interpolate N1 -> N2, then conv[3]
    knn_topk_kernel<<<cdivl(N2, TPB), TPB, 0, stream>>>(pos2, N2, pos1, N1, KI, KW);
    interp_gather_kernel<<<cdivl((long)N2 * 64, TPB), TPB, 0, stream>>>(X, KI, KW, T, N2);
    copy_kernel<<<cdivl((long)N2 * 64, TPB), TPB, 0, stream>>>(T, X, (long)N2 * 64);
    conv64(conv[3], e2, E2, N2, 1);

    // final conv (out=3, no relu) -> d_out
    {
        long n = (long)N2 * 64;
        copy_kernel<<<cdivl(n, TPB), TPB, 0, stream>>>(X, T, n);
        edge_scatter_kernel<<<cdivl((long)E2 * 16, TPB), TPB, 0, stream>>>(X, e2, T, E2);
        gin_mlp_final_kernel<<<cdivl(N2, TPB), TPB, 0, stream>>>(
            T, (float*)d_out,
            conv[4][0], conv[4][1], conv[4][2], conv[4][3],
            conv[4][4], conv[4][5], conv[4][6], conv[4][7], N2);
    }
}